// Node_GIN_33397665693790
// MI455X (gfx1250) — compile-verified
//
#include <hip/hip_runtime.h>

typedef __attribute__((ext_vector_type(2))) float v2f;
typedef __attribute__((ext_vector_type(8))) float v8f;

#define N_NODES 100000
#define N_EDGES 1600000
#define D_IN    64
#define D_HID   128
#define D_OUT   64

// ---------------------------------------------------------------------------
// Zero fill
// ---------------------------------------------------------------------------
__global__ __launch_bounds__(256) void zero_f32(float* __restrict__ p, long n) {
  long i = (long)blockIdx.x * blockDim.x + threadIdx.x;
  if (i < n) p[i] = 0.0f;
}

// ---------------------------------------------------------------------------
// Edge scatter-add: agg[dst[e]][f] += h[src[e]][f]
// One thread per (edge, feature). All lanes of a wave share one edge
// (dim >= 32): src/dst loads are wave-uniform, feature traffic is coalesced,
// atomics resolve in L2 (feature arrays are 51MB < 192MB L2).
// ---------------------------------------------------------------------------
__global__ __launch_bounds__(256) void scatter_add(const float* __restrict__ h,
                                                   const int*   __restrict__ src,
                                                   const int*   __restrict__ dst,
                                                   float*       __restrict__ agg,
                                                   int dim_log2) {
  long i = (long)blockIdx.x * blockDim.x + threadIdx.x;
  long total = (long)N_EDGES << dim_log2;
  if (i >= total) return;
  int dim = 1 << dim_log2;
  int e = (int)(i >> dim_log2);
  int f = (int)(i & (dim - 1));
  int s = src[e];
  int d = dst[e];
  unsafeAtomicAdd(&agg[(long)d * dim + f], h[(long)s * dim + f]);
}

// ---------------------------------------------------------------------------
// Fused GEMM: Y = act( (X [+ A]) @ W + bias ).
//   X: [nRows, K], A: optional add term (HAS_ADD), W: [K, M], bias: [M],
//   Y: [nRows, M].
// One wave owns a 16x64 output macro-tile: 4 v8f accumulators fed by
// v_wmma_f32_16x16x4_f32. W is staged in LDS *k-pair interleaved*:
//   WP[p][c] = (W[2p][c], W[2p+1][c])  as float2, row stride SP = M+16.
// So each B fragment is one aligned ds_load_b64 straight into the operand
// register pair (no v_mov reassembly), and 2*SP mod 64 == 32 keeps the two
// half-waves' bank footprints disjoint. Per K-step: 1 global_load_b64 for A
// (reused by 4 WMMAs), 4 b64 LDS reads, 4 WMMAs on independent accumulators.
// ---------------------------------------------------------------------------
template <int K, int M, bool RELU, bool HAS_ADD>
__global__ __launch_bounds__(256) void gemm_wmma_f32(const float* __restrict__ X,
                                                     const float* __restrict__ A,
                                                     const float* __restrict__ W,
                                                     const float* __restrict__ bias,
                                                     float*       __restrict__ Y,
                                                     int nRowTiles) {
  constexpr int SP = M + 16;                  // float2 stride per k-pair row
  constexpr int CW = 64;                      // macro-tile width (cols per wave)
  constexpr int NT = CW / 16;                 // 4 accumulators per wave
  constexpr int numColMacros = M / CW;

  __shared__ v2f Wl[(K / 2) * SP];
  for (int i = threadIdx.x; i < (K / 2) * M; i += 256) {
    int p = i / M, c = i - p * M;
    v2f w;
    w.x = W[(2 * p) * M + c];
    w.y = W[(2 * p + 1) * M + c];
    Wl[p * SP + c] = w;
  }
  __syncthreads();

  const int lane = threadIdx.x & 31;
  const int wave = threadIdx.x >> 5;
  const long numTiles = (long)nRowTiles * numColMacros;
  long tile = (long)blockIdx.x * 8 + wave;
  if (tile >= numTiles) return;               // wave-uniform: EXEC stays full

  const int rowTile  = (int)(tile / numColMacros);
  const int colMacro = (int)(tile - (long)rowTile * numColMacros);
  const int rowBase  = rowTile * 16;
  const int colBase  = colMacro * CW;

  // A fragment (16x4 f32): lane<16 -> row=lane, K={k0,k0+1}; lane>=16 -> K={k0+2,k0+3}
  const int aRow  = rowBase + (lane & 15);
  const int half  = lane >> 4;                // 0 or 1
  // B fragment (4x16 f32): lane&15 -> col, halves mirror A's K split
  const int bCol  = colBase + (lane & 15);

  const float* xrow = X + (long)aRow * K;
  const float* arow = HAS_ADD ? (A + (long)aRow * K) : nullptr;

  v8f acc[NT];
#pragma unroll
  for (int t = 0; t < NT; ++t) acc[t] = (v8f){0.f,0.f,0.f,0.f,0.f,0.f,0.f,0.f};

#pragma unroll
  for (int k0 = 0; k0 < K; k0 += 4) {
    const int ka = k0 + half * 2;             // even: k-pair base for this half-wave
    const int pk = (k0 >> 1) + half;          // k-pair row in Wl
    v2f a;
    a.x = xrow[ka];
    a.y = xrow[ka + 1];
    if constexpr (HAS_ADD) {
      a.x += arow[ka];
      a.y += arow[ka + 1];
    }
#pragma unroll
    for (int t = 0; t < NT; ++t) {
      v2f b = Wl[pk * SP + bCol + t * 16];    // single ds_load_b64 -> operand pair
      acc[t] = __builtin_amdgcn_wmma_f32_16x16x4_f32(false, a, false, b,
                                                     (short)0, acc[t], false, false);
    }
  }

  // C/D layout: VGPR r -> M = r (lanes 0-15) / r+8 (lanes 16-31), N = lane&15
  const int rBase = rowBase + half * 8;
#pragma unroll
  for (int t = 0; t < NT; ++t) {
    const int col = bCol + t * 16;
    const float bv = bias[col];
#pragma unroll
    for (int r = 0; r < 8; ++r) {
      float y = acc[t][r] + bv;
      if (RELU) y = fmaxf(y, 0.0f);
      Y[(long)(rBase + r) * M + col] = y;
    }
  }
}

// ---------------------------------------------------------------------------
// Launcher
// ---------------------------------------------------------------------------
extern "C" void kernel_launch(void* const* d_in, const int* in_sizes, int n_in,
                              void* d_out, int out_size, void* d_ws, size_t ws_size,
                              hipStream_t stream) {
  (void)in_sizes; (void)n_in; (void)out_size; (void)ws_size;

  const float* x    = (const float*)d_in[0];
  const int*   edge = (const int*)d_in[1];     // harness delivers integer inputs as int32
  const float* W1a  = (const float*)d_in[2];
  const float* b1a  = (const float*)d_in[3];
  const float* W1b  = (const float*)d_in[4];
  const float* b1b  = (const float*)d_in[5];
  const float* W2a  = (const float*)d_in[6];
  const float* b2a  = (const float*)d_in[7];
  const float* W2b  = (const float*)d_in[8];
  const float* b2b  = (const float*)d_in[9];
  const float* Wlin = (const float*)d_in[10];
  const float* blin = (const float*)d_in[11];

  const int* src = edge;               // edge_index[0, :]
  const int* dst = edge + N_EDGES;     // edge_index[1, :]

  float* agg = (float*)d_ws;                         // N*128 f32
  float* t   = agg + (size_t)N_NODES * D_HID;        // N*128 f32
  float* h   = t   + (size_t)N_NODES * D_HID;        // N*128 f32
  float* out = (float*)d_out;                        // N*64  f32

  const int rowTiles = N_NODES / 16;                 // 6250 exactly
  auto cdiv = [](long a, long b) { return (unsigned)((a + b - 1) / b); };

  // ----- conv1: agg = scatter(x) [dim 64] -----
  zero_f32<<<cdiv((long)N_NODES * D_IN, 256), 256, 0, stream>>>(agg, (long)N_NODES * D_IN);
  scatter_add<<<cdiv((long)N_EDGES * D_IN, 256), 256, 0, stream>>>(x, src, dst, agg, 6);
  // t = relu((x+agg) @ W1a + b1a)   [N,64]x[64,128]  -> 2 col-macros
  gemm_wmma_f32<D_IN, D_HID, true, true>
      <<<cdiv((long)rowTiles * 2, 8), 256, 0, stream>>>(x, agg, W1a, b1a, t, rowTiles);
  // h = relu(t @ W1b + b1b)         [N,128]x[128,128] (outer relu fused)
  gemm_wmma_f32<D_HID, D_HID, true, false>
      <<<cdiv((long)rowTiles * 2, 8), 256, 0, stream>>>(t, nullptr, W1b, b1b, h, rowTiles);

  // ----- conv2: agg = scatter(h) [dim 128] -----
  zero_f32<<<cdiv((long)N_NODES * D_HID, 256), 256, 0, stream>>>(agg, (long)N_NODES * D_HID);
  scatter_add<<<cdiv((long)N_EDGES * D_HID, 256), 256, 0, stream>>>(h, src, dst, agg, 7);
  // t = relu((h+agg) @ W2a + b2a)
  gemm_wmma_f32<D_HID, D_HID, true, true>
      <<<cdiv((long)rowTiles * 2, 8), 256, 0, stream>>>(h, agg, W2a, b2a, t, rowTiles);
  // agg <- relu(t @ W2b + b2b)   (agg buffer reused as h2)
  gemm_wmma_f32<D_HID, D_HID, true, false>
      <<<cdiv((long)rowTiles * 2, 8), 256, 0, stream>>>(t, nullptr, W2b, b2b, agg, rowTiles);

  // ----- final projection: out = h2 @ Wlin + blin   [N,128]x[128,64] -> 1 col-macro -----
  gemm_wmma_f32<D_HID, D_OUT, false, false>
      <<<cdiv((long)rowTiles * 1, 8), 256, 0, stream>>>(agg, nullptr, Wlin, blin, out, rowTiles);
}